// RelationalDepthLoss_77884936946326
// MI455X (gfx1250) — compile-verified
//
#include <hip/hip_runtime.h>
#include <hip/hip_bf16.h>

// ---- problem constants (match reference) ----
#define MARGIN_RANK 0.1f
#define MARGIN_OCC  0.3f
#define LAMBDA_OCC  1.5f
#define MIN_PIXELS  20.0f

constexpr int BB  = 4;
constexpr int NN  = 32;
constexpr int RR  = 64;
constexpr int HH  = 480;
constexpr int WW  = 640;
constexpr int HWC = HH * WW;                         // 307200
constexpr int PIX_PER_WAVE   = 1280;                 // 307200 / 1280 = 240 exact
constexpr int WAVES_PER_COMBO = HWC / PIX_PER_WAVE;  // 240
constexpr int WAVES_PER_BLOCK = 8;                   // 256 threads, wave32
constexpr int BLOCKS_X = WAVES_PER_COMBO / WAVES_PER_BLOCK; // 30
constexpr int COMBOS   = BB * 2;                     // 4 batches x 2 groups of 16 objects

typedef __attribute__((ext_vector_type(2))) float v2f;
typedef __attribute__((ext_vector_type(4))) float v4f;
typedef __attribute__((ext_vector_type(8))) float v8f;

// ---------------------------------------------------------------------------
// Kernel 0: zero the workspace (128 sums + 128 counts)
// ---------------------------------------------------------------------------
__global__ void RDL_zero_ws(float* __restrict__ ws) {
    ws[threadIdx.x] = 0.0f;   // launched with exactly 256 threads
}

// ---------------------------------------------------------------------------
// Kernel 1: per-(b,n) masked depth sums + pixel counts via V_WMMA_F32_16X16X4_F32
//   D = A x B + C, chained over K (pixels):
//     A[m][k] = (mask[b, objBase+m, pix_k] > 0.5) ? 1 : 0     (16 objects x 4 px)
//     B[k][0] = depth[b, pix_k]  -> D[:,0] = sum(m*depth)
//     B[k][1] = 1.0              -> D[:,1] = cnt ; other columns 0
//
// K-slot permutation trick: lane L (col = L&15, half = L>>4) loads ONE b128
// quad at  base = pStart + 4*half + p  (p += 8). The two WMMAs of the group
// use K-slots {2*half+0, 2*half+1} <- quad elements {x,y} then {z,w}.
// Since A and B draw the same pixel for the same K slot, the permuted sum
// equals the canonical one. Lanes m and m+16 jointly stream object m's row
// contiguously (full cacheline utilization); 1 vmem load per WMMA.
// ---------------------------------------------------------------------------
__global__ __launch_bounds__(256)
void RDL_mask_depth_wmma(const float* __restrict__ depth,
                         const float* __restrict__ masks,
                         float* __restrict__ ws_sum,
                         float* __restrict__ ws_cnt) {
    const int lane      = threadIdx.x & 31;
    const int waveInBlk = threadIdx.x >> 5;
    const int combo     = blockIdx.y;          // 0..7
    const int b         = combo >> 1;
    const int objBase   = (combo & 1) * 16;
    const int wave      = blockIdx.x * WAVES_PER_BLOCK + waveInBlk;
    const int pStart    = wave * PIX_PER_WAVE;

    const int col  = lane & 15;   // A-matrix row (object) and B/D column
    const int half = lane >> 4;   // selects K pair {0,1} vs {2,3}

    const float* __restrict__ dplane =
        depth + (size_t)b * (size_t)HWC;
    const float* __restrict__ mrow =
        masks + ((size_t)b * NN + (size_t)(objBase + col)) * (size_t)HWC;

    // B-column selectors, hoisted: col0 -> depth, col1 -> 1.0, else 0
    const float selD = (col == 0) ? 1.0f : 0.0f;
    const float selO = (col == 1) ? 1.0f : 0.0f;

    v8f c = {0.f, 0.f, 0.f, 0.f, 0.f, 0.f, 0.f, 0.f};

    const int base = pStart + 4 * half;        // 16-byte aligned
#pragma unroll 4
    for (int p = 0; p < PIX_PER_WAVE; p += 8) {
        const int off = base + p;
        const v4f mq = *(const v4f*)(mrow   + off);  // global_load_b128 (masks)
        const v4f dq = *(const v4f*)(dplane + off);  // global_load_b128 (depth, L2-hot)

        // WMMA #1: K pixels = quad elements {x, y} of each half
        v2f a1, b1;
        a1.x = (mq.x > 0.5f) ? 1.0f : 0.0f;
        a1.y = (mq.y > 0.5f) ? 1.0f : 0.0f;
        b1.x = __builtin_fmaf(dq.x, selD, selO);
        b1.y = __builtin_fmaf(dq.y, selD, selO);
        c = __builtin_amdgcn_wmma_f32_16x16x4_f32(
                false, a1, false, b1, (short)0, c, false, false);

        // WMMA #2: K pixels = quad elements {z, w} of each half
        v2f a2, b2;
        a2.x = (mq.z > 0.5f) ? 1.0f : 0.0f;
        a2.y = (mq.w > 0.5f) ? 1.0f : 0.0f;
        b2.x = __builtin_fmaf(dq.z, selD, selO);
        b2.y = __builtin_fmaf(dq.w, selD, selO);
        c = __builtin_amdgcn_wmma_f32_16x16x4_f32(
                false, a2, false, b2, (short)0, c, false, false);
    }

    // Extract: column 0 holds masked-depth sums, column 1 holds counts.
    // D layout: VGPR r, lanes 0-15 -> M=r, lanes 16-31 -> M=8+r.
    if (col <= 1) {
        float* __restrict__ dst = (col == 0) ? ws_sum : ws_cnt;
        const int objOff = b * NN + objBase + half * 8;
#pragma unroll
        for (int r = 0; r < 8; ++r)
            atomicAdd(&dst[objOff + r], c[r]);
    }
}

// ---------------------------------------------------------------------------
// Kernel 2: relation scoring + final scalar (one 256-thread block, B*R = 256)
// ---------------------------------------------------------------------------
__global__ __launch_bounds__(256)
void RDL_finalize(const int*   __restrict__ subj,
                  const int*   __restrict__ obj,
                  const int*   __restrict__ rel,
                  const float* __restrict__ conf,
                  const float* __restrict__ ws_sum,
                  const float* __restrict__ ws_cnt,
                  float* __restrict__ out) {
    __shared__ float s_tot[256];
    __shared__ float s_cnt[256];

    const int t = threadIdx.x;          // 0..255 == one relation
    const int b = t / RR;
    const int r = t - b * RR;
    const int idx = b * RR + r;

    const int si = subj[idx];
    const int oi = obj[idx];
    const int rt = rel[idx];

    const int iA = (rt == 1) ? oi : si;   // "behind" swaps roles
    const int iB = (rt == 1) ? si : oi;

    const float cntA = ws_cnt[b * NN + iA];
    const float cntB = ws_cnt[b * NN + iB];
    const float dA = ws_sum[b * NN + iA] / fmaxf(cntA, 1.0f);
    const float dB = ws_sum[b * NN + iB] / fmaxf(cntB, 1.0f);

    const bool valid = (cntA >= MIN_PIXELS) && (cntB >= MIN_PIXELS);
    const bool occ   = (rt == 2);
    const float margin = occ ? MARGIN_OCC : MARGIN_RANK;
    const float coeff  = occ ? LAMBDA_OCC : 1.0f;
    const float viol   = fmaxf(dA - dB + margin, 0.0f);
    const float w      = coeff * conf[idx] * (valid ? 1.0f : 0.0f);

    s_tot[t] = w * viol;
    s_cnt[t] = valid ? 1.0f : 0.0f;
    __syncthreads();

#pragma unroll
    for (int s = 128; s > 0; s >>= 1) {
        if (t < s) { s_tot[t] += s_tot[t + s]; s_cnt[t] += s_cnt[t + s]; }
        __syncthreads();
    }
    if (t == 0) {
        const float count = s_cnt[0];
        out[0] = (count > 0.0f) ? (s_tot[0] / fmaxf(count, 1.0f)) : 0.0f;
    }
}

// ---------------------------------------------------------------------------
extern "C" void kernel_launch(void* const* d_in, const int* in_sizes, int n_in,
                              void* d_out, int out_size, void* d_ws, size_t ws_size,
                              hipStream_t stream) {
    const float* depth = (const float*)d_in[0];   // (B,1,H,W) f32
    const float* masks = (const float*)d_in[1];   // (B,N,H,W) f32
    const int*   subj  = (const int*)  d_in[2];   // (B,R) i32
    const int*   obj   = (const int*)  d_in[3];   // (B,R) i32
    const int*   rel   = (const int*)  d_in[4];   // (B,R) i32
    const float* conf  = (const float*)d_in[5];   // (B,R) f32

    float* ws_sum = (float*)d_ws;                 // 128 floats
    float* ws_cnt = ws_sum + BB * NN;             // 128 floats

    RDL_zero_ws<<<1, 2 * BB * NN, 0, stream>>>(ws_sum);

    dim3 grid(BLOCKS_X, COMBOS);                  // (30, 8)
    RDL_mask_depth_wmma<<<grid, 256, 0, stream>>>(depth, masks, ws_sum, ws_cnt);

    RDL_finalize<<<1, 256, 0, stream>>>(subj, obj, rel, conf,
                                        ws_sum, ws_cnt, (float*)d_out);
}